// TokenUninterleaver_75436805587274
// MI455X (gfx1250) — compile-verified
//
#include <hip/hip_runtime.h>
#include <cstdint>

// Problem shape (fixed by the reference): B=4, S=4096, D=2048, 4 modalities.
#define BB 4
#define SS 4096
#define DD 2048
#define THREADS 256
#define TOK (SS / THREADS) // 16 tokens per thread in the index kernel

typedef float f4 __attribute__((ext_vector_type(4)));

// ---------------------------------------------------------------------------
// CDNA5 async global<->LDS DMA helpers (gfx1250, tracked on ASYNCcnt).
// Streaming workload: input read once, output written once and never re-read,
// and the 512 MiB output stream exceeds the 192 MB L2 -> use NT hints so the
// streams don't thrash the cache.
// ---------------------------------------------------------------------------
__device__ __forceinline__ void async_g2l_b128_nt(uint32_t lds_addr, uint64_t gaddr) {
  asm volatile("global_load_async_to_lds_b128 %0, %1, off th:TH_LOAD_NT"
               :: "v"(lds_addr), "v"(gaddr) : "memory");
}
__device__ __forceinline__ void async_l2g_b128_nt(uint64_t gaddr, uint32_t lds_addr) {
  asm volatile("global_store_async_from_lds_b128 %0, %1, off th:TH_STORE_NT"
               :: "v"(gaddr), "v"(lds_addr) : "memory");
}
__device__ __forceinline__ void wait_async0() {
  asm volatile("s_wait_asynccnt 0x0" ::: "memory");
}

// ---------------------------------------------------------------------------
// Kernel A: per batch row, stable-compaction prefix sums over the modality map.
// Produces src_idx[m][b][j] = source token s for output slot j, and lengths.
// Also writes the 4x4 length outputs (as float, matching d_out's dtype).
// Tiny: 4 blocks x 256 threads over 16K map entries.
// ---------------------------------------------------------------------------
__global__ void tu_index_kernel(const int* __restrict__ mmap,
                                int* __restrict__ src_idx,
                                int* __restrict__ lens,
                                float* __restrict__ out) {
  const int b = blockIdx.x;
  const int t = threadIdx.x;
  __shared__ int cnt[4][THREADS];
  __shared__ int pre[4][THREADS];
  __shared__ int tot[4];

  int mods[TOK];
  int local[4] = {0, 0, 0, 0};
  const int srow = t * TOK;
  const int* row = mmap + b * SS;
#pragma unroll
  for (int i = 0; i < TOK; ++i) {
    int m = row[srow + i] & 3;
    mods[i] = m;
    local[m]++;
  }
#pragma unroll
  for (int m = 0; m < 4; ++m) cnt[m][t] = local[m];
  __syncthreads();

  if (t == 0) {
    // Sequential exclusive scan over 256 thread-counts per modality.
    // 1024 adds on 4 blocks total -> negligible vs the 640 MiB copy.
    for (int m = 0; m < 4; ++m) {
      int acc = 0;
      for (int i = 0; i < THREADS; ++i) { pre[m][i] = acc; acc += cnt[m][i]; }
      tot[m] = acc;
    }
  }
  __syncthreads();

  int pos[4] = {pre[0][t], pre[1][t], pre[2][t], pre[3][t]};
#pragma unroll
  for (int i = 0; i < TOK; ++i) {
    int m = mods[i];
    src_idx[(m * BB + b) * SS + pos[m]] = srow + i;
    pos[m]++;
  }

  if (t < 4) {
    lens[t * BB + b] = tot[t];
    // d_out layout per modality m: [B*S*D data][B lengths], concatenated.
    size_t len_off = (size_t)t * ((size_t)BB * SS * DD + BB) + (size_t)BB * SS * DD;
    out[len_off + b] = (float)tot[t];
  }
}

// ---------------------------------------------------------------------------
// Kernel B: one block per OUTPUT row (m, b, j). 4*B*S = 65536 blocks.
// Copy rows: stream 8 KiB global->LDS->global via the async DMA path
// (ASYNCcnt-tracked, no VGPR data movement, NT cache policy). Zero rows:
// direct non-temporal b128 stores. All writes are fully coalesced contiguous
// rows -> bandwidth-bound. 8 KiB LDS/block -> 32 workgroups/WGP resident, so
// the asynccnt waits are hidden by multi-wave occupancy.
// ---------------------------------------------------------------------------
__global__ void tu_scatter_kernel(const float* __restrict__ tokens,
                                  const int* __restrict__ src_idx,
                                  const int* __restrict__ lens,
                                  float* __restrict__ out) {
  __shared__ __align__(16) float smem[DD]; // 8 KiB row staging
  const int rid = blockIdx.x;
  const int j = rid & (SS - 1);         // 12 bits
  const int b = (rid >> 12) & (BB - 1); // 2 bits
  const int m = rid >> 14;              // 2 bits
  const int t = threadIdx.x;

  const int len = lens[m * BB + b];
  float* dst = out + (size_t)m * ((size_t)BB * SS * DD + BB)
                   + ((size_t)b * SS + j) * DD;

  if (j < len) {
    const int s = src_idx[(m * BB + b) * SS + j];
    const float* src = tokens + ((size_t)b * SS + s) * DD;

    const uint32_t lbase = (uint32_t)(size_t)&smem[0]; // LDS byte offset (low 32 bits)
    const uint64_t gsrc = (uint64_t)src;
    const uint64_t gdst = (uint64_t)dst;
    const uint32_t off0 = (uint32_t)t * 16u;           // 16 B per lane per op
    const uint32_t off1 = off0 + THREADS * 16u;

    // 2 async loads in flight per thread, one wait, 2 async stores, done.
    async_g2l_b128_nt(lbase + off0, gsrc + off0);
    async_g2l_b128_nt(lbase + off1, gsrc + off1);
    wait_async0();                    // loads landed in this thread's LDS slice
    async_l2g_b128_nt(gdst + off0, lbase + off0);
    async_l2g_b128_nt(gdst + off1, lbase + off1);
    wait_async0();
  } else {
    // Zero tail: no LDS round trip needed for a constant; NT stores keep the
    // 384 MiB of zero-fill out of L2.
    f4 z = (f4)0.f;
    f4* d4 = (f4*)dst;
    __builtin_nontemporal_store(z, d4 + t);
    __builtin_nontemporal_store(z, d4 + t + THREADS);
  }
}

// ---------------------------------------------------------------------------
// Launch: index build (ordered before) the gather/zero-fill streamer.
// Workspace: src_idx = 4*B*S ints (256 KiB), lens = 16 ints.
// ---------------------------------------------------------------------------
extern "C" void kernel_launch(void* const* d_in, const int* in_sizes, int n_in,
                              void* d_out, int out_size, void* d_ws, size_t ws_size,
                              hipStream_t stream) {
  const float* tokens = (const float*)d_in[0];
  const int* mmap = (const int*)d_in[1];
  float* out = (float*)d_out;

  int* src_idx = (int*)d_ws;
  int* lens = src_idx + 4 * BB * SS;

  tu_index_kernel<<<BB, THREADS, 0, stream>>>(mmap, src_idx, lens, out);
  tu_scatter_kernel<<<4 * BB * SS, THREADS, 0, stream>>>(tokens, src_idx, lens, out);
}